// GNN_70978629533716
// MI455X (gfx1250) — compile-verified
//
#include <hip/hip_runtime.h>
#include <hip/hip_bf16.h>

typedef float v2f __attribute__((ext_vector_type(2)));
typedef float v8f __attribute__((ext_vector_type(8)));

#define LRELU(v) ((v) > 0.0f ? (v) : 0.2f * (v))
#define RELU(v)  ((v) > 0.0f ? (v) : 0.0f)

// Monotone float <-> uint mapping so unsigned atomicMax implements float max.
__device__ __forceinline__ unsigned fenc(float x) {
    unsigned u = __float_as_uint(x);
    return (u & 0x80000000u) ? ~u : (u | 0x80000000u);
}
__device__ __forceinline__ float fdec(unsigned e) {
    return __uint_as_float((e & 0x80000000u) ? (e & 0x7fffffffu) : ~e);
}

// ---------------------------------------------------------------- zero fill
__global__ void zero_f32(float* __restrict__ p, size_t n) {
    size_t i = (size_t)blockIdx.x * blockDim.x + threadIdx.x;
    size_t stride = (size_t)gridDim.x * blockDim.x;
    for (; i < n; i += stride) p[i] = 0.0f;
}

// ---------------------------------------------------------------- front MLP
// x[N,1] -> 16 -> 32 -> 64 (ReLU each), one thread per node.
__global__ void mlp_kernel(const float* __restrict__ x,
                           const float* __restrict__ w1, const float* __restrict__ b1,
                           const float* __restrict__ w2, const float* __restrict__ b2,
                           const float* __restrict__ w3, const float* __restrict__ b3,
                           float* __restrict__ out, int N) {
    int n = blockIdx.x * blockDim.x + threadIdx.x;
    if (n >= N) return;
    float xv = x[n];
    float h1[16];
#pragma unroll
    for (int j = 0; j < 16; ++j) h1[j] = RELU(xv * w1[j] + b1[j]);
    float h2[32];
#pragma unroll
    for (int j = 0; j < 32; ++j) {
        float acc = b2[j];
#pragma unroll
        for (int k = 0; k < 16; ++k) acc += h1[k] * w2[j * 16 + k];
        h2[j] = RELU(acc);
    }
    float* o = out + (size_t)n * 64;
    for (int j = 0; j < 64; ++j) {
        float acc = b3[j];
#pragma unroll
        for (int k = 0; k < 32; ++k) acc += h2[k] * w3[j * 32 + k];
        o[j] = RELU(acc);
    }
}

// ---------------------------------------------------------------- WMMA GEMM
// B[NPAD,OD] = A[NPAD,ID] @ W[OD,ID]^T  using V_WMMA_F32_16X16X4_F32.
// Block = 256 threads = 8 waves; wave w owns rows [bx*128+16w, +16),
// cols [by*32, +32) (two 16x16 accumulators, A fragment reused).
__global__ void gemm_wmma_f32(const float* __restrict__ A,
                              const float* __restrict__ W,
                              float* __restrict__ B, int ID, int OD) {
    int wave = threadIdx.x >> 5;
    int lane = threadIdx.x & 31;
    int half = lane >> 4;        // K sub-block select per ISA layout
    int lrow = lane & 15;
    int m0 = blockIdx.x * 128 + wave * 16;
    int o0 = blockIdx.y * 32;

    const float* arow  = A + (size_t)(m0 + lrow) * ID + 2 * half;
    const float* wrow0 = W + (size_t)(o0 + lrow) * ID + 2 * half;
    const float* wrow1 = W + (size_t)(o0 + 16 + lrow) * ID + 2 * half;

    v8f c0 = {}; v8f c1 = {};
    for (int k = 0; k < ID; k += 4) {
        v2f a  = *(const v2f*)(arow + k);
        v2f b0 = *(const v2f*)(wrow0 + k);
        v2f b1 = *(const v2f*)(wrow1 + k);
        c0 = __builtin_amdgcn_wmma_f32_16x16x4_f32(false, a, false, b0,
                                                   (short)0, c0, false, false);
        c1 = __builtin_amdgcn_wmma_f32_16x16x4_f32(false, a, false, b1,
                                                   (short)0, c1, false, false);
    }
    // D layout: lane<16 -> (M = r, O = lane); lane>=16 -> (M = 8+r, O = lane-16)
    float* out = B + (size_t)(m0 + 8 * half) * OD + o0 + lrow;
#pragma unroll
    for (int r = 0; r < 8; ++r) {
        out[(size_t)r * OD]      = c0[r];
        out[(size_t)r * OD + 16] = c1[r];
    }
}

// ------------------------------------------------------- per-node alpha init
// One wave per node: alpha_src/dst dots, seed segment-max with the self-loop
// logit, zero den and the aggregation accumulator row.
__global__ void alpha_init(const float* __restrict__ H,
                           const float* __restrict__ avs, const float* __restrict__ avd,
                           float* __restrict__ as_, float* __restrict__ ad_,
                           unsigned* __restrict__ mEnc, float* __restrict__ den,
                           float* __restrict__ agg, int OD, int N) {
    int lane = threadIdx.x & 31;
    int n = blockIdx.x * (blockDim.x >> 5) + (threadIdx.x >> 5);
    if (n >= N) return;
    const float* h = H + (size_t)n * OD;
    float s = 0.0f, d = 0.0f;
    for (int f = lane; f < OD; f += 32) {
        float v = h[f];
        s += v * avs[f];
        d += v * avd[f];
    }
#pragma unroll
    for (int off = 16; off > 0; off >>= 1) {
        s += __shfl_xor(s, off);
        d += __shfl_xor(d, off);
    }
    float* ag = agg + (size_t)n * OD;
    for (int f = lane; f < OD; f += 32) ag[f] = 0.0f;
    if (lane == 0) {
        as_[n] = s;
        ad_[n] = d;
        float e = LRELU(s + d);          // self-loop logit seeds the max
        mEnc[n] = fenc(e);
        den[n]  = 0.0f;
    }
}

// ------------------------------------------------------------ edge pass 1
__global__ void edge_max(const int* __restrict__ src, const int* __restrict__ dst,
                         const float* __restrict__ as_, const float* __restrict__ ad_,
                         unsigned* __restrict__ mEnc, int E) {
    int e = blockIdx.x * blockDim.x + threadIdx.x;
    if (e >= E) return;
    int s = src[e], d = dst[e];
    float v = LRELU(as_[s] + ad_[d]);
    atomicMax(&mEnc[d], fenc(v));
}

// ------------------------------------------------------------ edge pass 2
// One wave per edge: ex = exp(e - m[dst]); den[dst] += ex;
// agg[dst,:] += h[src,:] * ex   (f32 atomics hit L2 — accumulator fits).
__global__ void edge_accum(const int* __restrict__ src, const int* __restrict__ dst,
                           const float* __restrict__ H,
                           const float* __restrict__ as_, const float* __restrict__ ad_,
                           const unsigned* __restrict__ mEnc,
                           float* __restrict__ den, float* __restrict__ agg,
                           int OD, int E) {
    int lane = threadIdx.x & 31;
    int e = blockIdx.x * (blockDim.x >> 5) + (threadIdx.x >> 5);
    if (e >= E) return;
    int s = src[e], d = dst[e];
    float v  = LRELU(as_[s] + ad_[d]);
    float ex = __expf(v - fdec(mEnc[d]));
    if (lane == 0) atomicAdd(&den[d], ex);
    const float* hs = H + (size_t)s * OD;
    float* ag = agg + (size_t)d * OD;
    for (int f = lane; f < OD; f += 32) atomicAdd(&ag[f], hs[f] * ex);
}

// ------------------------------------------------------------- finalize
// One wave per node: add self-loop term, normalize, + bias, ReLU (in place).
__global__ void finalize_gat(float* __restrict__ agg, const float* __restrict__ H,
                             const float* __restrict__ as_, const float* __restrict__ ad_,
                             const unsigned* __restrict__ mEnc, const float* __restrict__ den,
                             const float* __restrict__ bias, int OD, int N) {
    int lane = threadIdx.x & 31;
    int n = blockIdx.x * (blockDim.x >> 5) + (threadIdx.x >> 5);
    if (n >= N) return;
    float e  = LRELU(as_[n] + ad_[n]);
    float ex = __expf(e - fdec(mEnc[n]));
    float inv = 1.0f / (den[n] + ex);
    const float* hn = H + (size_t)n * OD;
    float* ag = agg + (size_t)n * OD;
    for (int f = lane; f < OD; f += 32) {
        float o = (ag[f] + hn[f] * ex) * inv + bias[f];
        ag[f] = RELU(o);
    }
}

// ------------------------------------------------------------- pooling
__global__ void pool_accum(const float* __restrict__ H, const int* __restrict__ batch,
                           float* __restrict__ gsum, float* __restrict__ gcnt, int N) {
    int lane = threadIdx.x & 31;
    int n = blockIdx.x * (blockDim.x >> 5) + (threadIdx.x >> 5);
    if (n >= N) return;
    int g = batch[n];
    atomicAdd(&gsum[(size_t)g * 32 + lane], H[(size_t)n * 32 + lane]);
    if (lane == 0) atomicAdd(&gcnt[g], 1.0f);
}

__global__ void pool_final(const float* __restrict__ gsum, const float* __restrict__ gcnt,
                           const float* __restrict__ lw, const float* __restrict__ lb,
                           float* __restrict__ out, int G) {
    int g = blockIdx.x * blockDim.x + threadIdx.x;
    if (g >= G) return;
    float inv = 1.0f / fmaxf(gcnt[g], 1.0f);
    float z = lb[0];
    const float* s = gsum + (size_t)g * 32;
#pragma unroll
    for (int k = 0; k < 32; ++k) z += (s[k] * inv) * lw[k];
    out[g] = 1.0f / (1.0f + __expf(-z));
}

// ----------------------------------------------------------------- launch
extern "C" void kernel_launch(void* const* d_in, const int* in_sizes, int n_in,
                              void* d_out, int out_size, void* d_ws, size_t ws_size,
                              hipStream_t stream) {
    const float* x     = (const float*)d_in[0];
    const int*   ei    = (const int*)  d_in[1];
    const int*   batch = (const int*)  d_in[2];
    const float* w1 = (const float*)d_in[3], *b1 = (const float*)d_in[4];
    const float* w2 = (const float*)d_in[5], *b2 = (const float*)d_in[6];
    const float* w3 = (const float*)d_in[7], *b3 = (const float*)d_in[8];
    const float* lw = (const float*)d_in[29], *lb = (const float*)d_in[30];

    const int N = in_sizes[0];          // x is [N,1]
    const int E = in_sizes[1] / 2;      // edge_index is [2,E]
    const int G = out_size;             // output [G,1]
    const int NPAD = ((N + 127) / 128) * 128;

    // Workspace carve-up (floats): A | B | as | ad | mEnc | den | gsum | gcnt
    size_t big = (size_t)NPAD * 256;
    float*    A    = (float*)d_ws;
    float*    B    = A + big;
    float*    as_  = B + big;
    float*    ad_  = as_ + N;
    unsigned* mE   = (unsigned*)(ad_ + N);
    float*    den  = (float*)(mE + N);
    float*    gsum = den + N;
    float*    gcnt = gsum + (size_t)G * 32;   // gcnt contiguous after gsum

    const int* src = ei;
    const int* dst = ei + E;

    // Deterministic init: zero both big buffers so padded GEMM rows read zeros.
    zero_f32<<<2048, 256, 0, stream>>>(A, 2 * big);

    // Front MLP -> A holds h[N,64]
    mlp_kernel<<<(N + 255) / 256, 256, 0, stream>>>(x, w1, b1, w2, b2, w3, b3, A, N);

    struct Layer { int W, av_s, av_d, bias, id, od; };
    const Layer L[5] = {
        { 9, 10, 11, 12,  64, 128},
        {13, 14, 15, 16, 128, 256},
        {17, 18, 19, 20, 256, 128},
        {21, 22, 23, 24, 128,  64},
        {25, 26, 27, 28,  64,  32},
    };

    const int nodeWaves = (N + 7) / 8;   // 8 waves per 256-thread block
    const int edgeWaves = (E + 7) / 8;

    for (int l = 0; l < 5; ++l) {
        const float* W   = (const float*)d_in[L[l].W];
        const float* avs = (const float*)d_in[L[l].av_s];
        const float* avd = (const float*)d_in[L[l].av_d];
        const float* bia = (const float*)d_in[L[l].bias];
        const int ID = L[l].id, OD = L[l].od;

        gemm_wmma_f32<<<dim3(NPAD / 128, OD / 32), 256, 0, stream>>>(A, W, B, ID, OD);
        alpha_init   <<<nodeWaves, 256, 0, stream>>>(B, avs, avd, as_, ad_, mE, den, A, OD, N);
        edge_max     <<<(E + 255) / 256, 256, 0, stream>>>(src, dst, as_, ad_, mE, E);
        edge_accum   <<<edgeWaves, 256, 0, stream>>>(src, dst, B, as_, ad_, mE, den, A, OD, E);
        finalize_gat <<<nodeWaves, 256, 0, stream>>>(A, B, as_, ad_, mE, den, bia, OD, N);
    }

    // Mean pool per graph + linear(32->1) + sigmoid
    zero_f32  <<<(G * 33 + 255) / 256, 256, 0, stream>>>(gsum, (size_t)G * 33);
    pool_accum<<<nodeWaves, 256, 0, stream>>>(A, batch, gsum, gcnt, N);
    pool_final<<<(G + 255) / 256, 256, 0, stream>>>(gsum, gcnt, lw, lb, (float*)d_out, G);
}